// FARGAN_80745385165059
// MI455X (gfx1250) — compile-verified
//
#include <hip/hip_runtime.h>

typedef __attribute__((ext_vector_type(2))) float v2f;
typedef __attribute__((ext_vector_type(8))) float v8f;

#define NTHREADS 256
#define NWAVES 8

// ---------------- LDS layout (float offsets), heavily aliased ----------------
#define OFF_RING  0                    // 16x512 ring buffer of prev samples (8192)
#define OFF_SUBS  8192                 // 4 x (16x128) subframe conditioning (8192)
#define OFF_T1    OFF_SUBS             // alias: t1 (16x256) during frame conditioning
#define OFF_SFW   16384                // sfw state 16x128 (2048)
#define OFF_S1    18432                // GRU states 16x64 each (contiguous 3072)
#define OFF_S2    19456
#define OFF_S3    20480
#define OFF_AIN   21504                // fw input 16x388 (6208)
#define OFF_X     OFF_AIN              // alias: x (16x256) during frame conditioning
#define OFF_GRUIN OFF_AIN              // alias: GRU input 16x128
#define OFF_SD    (OFF_AIN + 2048)     // alias: sd/so 16x128 (2048)
#define OFF_SKIP  27712                // skip 16x320 = [o1|o2|o3|fw|prev_sub] (5120)
#define OFF_GI    32832                // GRU gi 16x192 (3072)
#define OFF_GATE  OFF_GI               // alias: Wsg gate 16x128
#define OFF_T2    OFF_GI               // alias: t2 (16x256) spans into GH
#define OFF_GH    35904                // GRU gh 16x192 (3072)
#define OFF_TG    OFF_GH               // alias: GLU gate logits 16x64
#define OFF_T64   (OFF_GH + 1024)      // alias: tanh temp / out 16x64
#define OFF_PER   38976                // per-row period (16 ints)
#define SMEM_FLOATS 38992              // ~156 KB dynamic LDS per workgroup

__device__ __forceinline__ float frcp(float x) { return __builtin_amdgcn_rcpf(x); }
__device__ __forceinline__ float fsigmoid(float x) {
  return frcp(1.0f + __expf(-x));                 // v_exp + v_rcp
}
__device__ __forceinline__ float ftanh(float x) {
  return 1.0f - 2.0f * frcp(__expf(2.0f * x) + 1.0f); // v_exp + v_rcp + fma
}

__device__ __forceinline__ v8f wmma4(v2f a, v2f b, v8f c) {
  return __builtin_amdgcn_wmma_f32_16x16x4_f32(false, a, false, b, (short)0, c,
                                               false, false);
}

template <int ACT>
__device__ __forceinline__ void store_tile(float* __restrict__ out, int ldo,
                                           int tile, int m, int h, const v8f& acc) {
  float* orow = out + tile * 16 + m;
#pragma unroll
  for (int r = 0; r < 8; ++r) {
    float v = acc[r];
    if (ACT == 1) v = ftanh(v);
    orow[(r + 8 * h) * ldo] = v;
  }
}

// D(16xN) = act( A(16xK, LDS) @ W^T ), W is [N x K] row-major in global (L2-hot).
// V_WMMA_F32_16X16X4_F32, wave32 fragment layout per CDNA5 ISA:
//   lane = m + 16*h :  A frag = A[m][k+2h..+1] (ds b64), B frag = W[n=m][k+2h..+1]
//   D: vgpr r -> row (r+8h), col m.
// Two N-tiles per wave (independent accumulator chains, shared A loads) when
// enough tiles exist; the pair/single split is wave-uniform so EXEC stays full.
template <int ACT> // 0 = none, 1 = tanh
__device__ __forceinline__ void gemm16(float* __restrict__ out, int ldo,
                                       const float* __restrict__ A, int lda,
                                       const float* __restrict__ W, int N, int K) {
  const int lane = threadIdx.x & 31;
  const int wave = threadIdx.x >> 5;
  const int m = lane & 15;
  const int h = lane >> 4;
  const float* arow = A + m * lda + 2 * h;
  const int ntiles = N >> 4;
  for (int t = wave; t < ntiles; t += 2 * NWAVES) {
    const float* wrow0 = W + (size_t)(t * 16 + m) * K + 2 * h;
    if (t + NWAVES < ntiles) {
      const float* wrow1 = wrow0 + (size_t)NWAVES * 16 * K;
      v8f acc0 = {0.f, 0.f, 0.f, 0.f, 0.f, 0.f, 0.f, 0.f};
      v8f acc1 = {0.f, 0.f, 0.f, 0.f, 0.f, 0.f, 0.f, 0.f};
#pragma unroll 4
      for (int k = 0; k < K; k += 4) {
        v2f a = *(const v2f*)(arow + k);
        acc0 = wmma4(a, *(const v2f*)(wrow0 + k), acc0);
        acc1 = wmma4(a, *(const v2f*)(wrow1 + k), acc1);
      }
      store_tile<ACT>(out, ldo, t, m, h, acc0);
      store_tile<ACT>(out, ldo, t + NWAVES, m, h, acc1);
    } else {
      v8f acc = {0.f, 0.f, 0.f, 0.f, 0.f, 0.f, 0.f, 0.f};
#pragma unroll 4
      for (int k = 0; k < K; k += 4)
        acc = wmma4(*(const v2f*)(arow + k), *(const v2f*)(wrow0 + k), acc);
      store_tile<ACT>(out, ldo, t, m, h, acc);
    }
  }
}

// Permuted store for c = tanh(t2 @ Wc3^T): col n -> subs[n&3][b][n>>2]
__device__ __forceinline__ void store_subs(float* __restrict__ subs, int tile,
                                           int m, int h, const v8f& acc) {
  const int n = tile * 16 + m;
  float* dst = subs + (n & 3) * 2048 + (n >> 2);
#pragma unroll
  for (int r = 0; r < 8; ++r) dst[(r + 8 * h) * 128] = ftanh(acc[r]);
}

__device__ __forceinline__ void gemm16_subs(float* __restrict__ subs,
                                            const float* __restrict__ A, int lda,
                                            const float* __restrict__ W, int N, int K) {
  const int lane = threadIdx.x & 31;
  const int wave = threadIdx.x >> 5;
  const int m = lane & 15;
  const int h = lane >> 4;
  const float* arow = A + m * lda + 2 * h;
  const int ntiles = N >> 4;
  for (int t = wave; t < ntiles; t += 2 * NWAVES) {
    const float* wrow0 = W + (size_t)(t * 16 + m) * K + 2 * h;
    const float* wrow1 = wrow0 + (size_t)NWAVES * 16 * K; // N=512: pair always valid
    v8f acc0 = {0.f, 0.f, 0.f, 0.f, 0.f, 0.f, 0.f, 0.f};
    v8f acc1 = {0.f, 0.f, 0.f, 0.f, 0.f, 0.f, 0.f, 0.f};
#pragma unroll 4
    for (int k = 0; k < K; k += 4) {
      v2f a = *(const v2f*)(arow + k);
      acc0 = wmma4(a, *(const v2f*)(wrow0 + k), acc0);
      acc1 = wmma4(a, *(const v2f*)(wrow1 + k), acc1);
    }
    store_subs(subs, t, m, h, acc0);
    store_subs(subs, t + NWAVES, m, h, acc1);
  }
}

// One GRU + GLU stage. gruin (16x128) = [x_part | prev_sub]; st = state (16x64).
__device__ __forceinline__ void gru_glu_step(float* __restrict__ sm,
                                             float* __restrict__ st,
                                             const float* __restrict__ Wih,
                                             const float* __restrict__ Whh,
                                             const float* __restrict__ Wg,
                                             int skip_off, int tid) {
  gemm16<0>(sm + OFF_GI, 192, sm + OFF_GRUIN, 128, Wih, 192, 128);
  gemm16<0>(sm + OFF_GH, 192, st, 64, Whh, 192, 64);
  __syncthreads();
  for (int i = tid; i < 16 * 64; i += NTHREADS) {
    int b = i >> 6, j = i & 63;
    const float* gi = sm + OFF_GI + b * 192;
    const float* gh = sm + OFF_GH + b * 192;
    float r = fsigmoid(gi[j] + gh[j]);
    float z = fsigmoid(gi[64 + j] + gh[64 + j]);
    float n = ftanh(gi[128 + j] + r * gh[128 + j]);
    st[i] = (1.0f - z) * n + z * st[i];
  }
  __syncthreads();
  gemm16<0>(sm + OFF_TG, 64, st, 64, Wg, 64, 64);
  __syncthreads();
  for (int i = tid; i < 16 * 64; i += NTHREADS) {
    int b = i >> 6, j = i & 63;
    float o = st[i] * fsigmoid(sm[OFF_TG + i]);
    sm[OFF_SKIP + b * 320 + skip_off + j] = o;
    sm[OFF_GRUIN + b * 128 + j] = o; // input for next GRU stage
  }
  __syncthreads();
}

__global__ __launch_bounds__(NTHREADS, 1) void fargan_persistent(
    const float* __restrict__ features, const float* __restrict__ gfeat,
    const float* __restrict__ prev0, const float* __restrict__ Wc1,
    const float* __restrict__ Wc2, const float* __restrict__ Wc3,
    const float* __restrict__ Wfw, const float* __restrict__ Wfwg,
    const float* __restrict__ Wih1, const float* __restrict__ Whh1,
    const float* __restrict__ Wih2, const float* __restrict__ Whh2,
    const float* __restrict__ Wih3, const float* __restrict__ Whh3,
    const float* __restrict__ Wg1, const float* __restrict__ Wg2,
    const float* __restrict__ Wg3, const float* __restrict__ Wsg,
    const float* __restrict__ Wsd, const float* __restrict__ Wout,
    float* __restrict__ outp) {
  extern __shared__ float sm[];
  int* perbuf = (int*)(sm + OFF_PER);
  const int b0 = blockIdx.x * 16;
  const int tid = threadIdx.x;

  // ---- init: prev samples -> ring, zero states ----
  for (int i = tid; i < 16 * 512; i += NTHREADS)
    sm[OFF_RING + i] = prev0[(size_t)b0 * 512 + i];
  for (int i = tid; i < 2048; i += NTHREADS) sm[OFF_SFW + i] = 0.0f;
  for (int i = tid; i < 3072; i += NTHREADS) sm[OFF_S1 + i] = 0.0f; // s1,s2,s3
  __syncthreads();

  int head = 0;
  for (int t = 0; t < 100; ++t) {
    // ---- frame conditioning: x = [feat | gfeat] ----
    for (int i = tid; i < 16 * 256; i += NTHREADS) {
      int b = i >> 8, j = i & 255;
      float v = (j < 192) ? features[(size_t)(b0 + b) * 19300 + j * 100 + t]
                          : gfeat[(size_t)(b0 + b) * 64 + (j - 192)];
      sm[OFF_X + i] = v;
    }
    if (tid < 16)
      perbuf[tid] = __float2int_rn(features[(size_t)(b0 + tid) * 19300 + 19200 + t]);
    __syncthreads();
    gemm16<1>(sm + OFF_T1, 256, sm + OFF_X, 256, Wc1, 256, 256);
    __syncthreads();
    gemm16<1>(sm + OFF_T2, 256, sm + OFF_T1, 256, Wc2, 256, 256);
    __syncthreads();
    gemm16_subs(sm + OFF_SUBS, sm + OFF_T2, 256, Wc3, 512, 256);
    __syncthreads();

    for (int k = 0; k < 4; ++k) {
      const float* subk = sm + OFF_SUBS + k * 2048;
      // ---- build fw input: [feat2s(128) | prev_sub(64) | lookback(68) | sfw(128)] ----
      for (int i = tid; i < 16 * 388; i += NTHREADS) {
        int b = i / 388, j = i - b * 388;
        float v;
        if (j < 128) v = subk[b * 128 + j];
        else if (j < 192)
          v = sm[OFF_RING + b * 512 + ((head + 448 + (j - 128)) & 511)];
        else if (j < 260) {
          int p = perbuf[b];
          int idx = 512 - p + (j - 192) - 2;
          if (idx >= 512) idx -= p;
          v = sm[OFF_RING + b * 512 + ((head + idx) & 511)];
        } else v = sm[OFF_SFW + b * 128 + (j - 260)];
        sm[OFF_AIN + i] = v;
      }
      __syncthreads();
      gemm16<1>(sm + OFF_T64, 64, sm + OFF_AIN, 388, Wfw, 64, 388);
      __syncthreads();
      gemm16<0>(sm + OFF_TG, 64, sm + OFF_T64, 64, Wfwg, 64, 64);
      __syncthreads();
      // fw = t*sigmoid(g); seed gruin and skip regions
      for (int i = tid; i < 16 * 64; i += NTHREADS) {
        int b = i >> 6, j = i & 63;
        float f = sm[OFF_T64 + i] * fsigmoid(sm[OFF_TG + i]);
        sm[OFF_SKIP + b * 320 + 192 + j] = f;
        sm[OFF_GRUIN + b * 128 + j] = f;
        float ps = sm[OFF_RING + b * 512 + ((head + 448 + j) & 511)];
        sm[OFF_GRUIN + b * 128 + 64 + j] = ps;
        sm[OFF_SKIP + b * 320 + 256 + j] = ps;
      }
      __syncthreads();

      gru_glu_step(sm, sm + OFF_S1, Wih1, Whh1, Wg1, 0, tid);
      gru_glu_step(sm, sm + OFF_S2, Wih2, Whh2, Wg2, 64, tid);
      gru_glu_step(sm, sm + OFF_S3, Wih3, Whh3, Wg3, 128, tid);

      // ---- skip dense + GLU + out ----
      gemm16<1>(sm + OFF_SD, 128, sm + OFF_SKIP, 320, Wsd, 128, 320);
      __syncthreads();
      gemm16<0>(sm + OFF_GATE, 128, sm + OFF_SD, 128, Wsg, 128, 128);
      __syncthreads();
      for (int i = tid; i < 16 * 128; i += NTHREADS)
        sm[OFF_SD + i] = sm[OFF_SD + i] * fsigmoid(sm[OFF_GATE + i]);
      __syncthreads();
      gemm16<1>(sm + OFF_T64, 64, sm + OFF_SD, 128, Wout, 64, 128);
      __syncthreads();

      // ---- emit, advance ring, carry sfw = feat2s ----
      head = (head + 64) & 511;
      for (int i = tid; i < 16 * 64; i += NTHREADS) {
        int b = i >> 6, j = i & 63;
        float o = sm[OFF_T64 + i];
        outp[(size_t)(b0 + b) * 25600 + t * 256 + k * 64 + j] = o;
        sm[OFF_RING + b * 512 + ((head + 448 + j) & 511)] = o;
      }
      for (int i = tid; i < 16 * 128; i += NTHREADS) sm[OFF_SFW + i] = subk[i];
      __syncthreads();
    }
  }
}

extern "C" void kernel_launch(void* const* d_in, const int* in_sizes, int n_in,
                              void* d_out, int out_size, void* d_ws, size_t ws_size,
                              hipStream_t stream) {
  (void)in_sizes; (void)n_in; (void)out_size; (void)d_ws; (void)ws_size;
  const float* features = (const float*)d_in[0];
  const float* gfeat    = (const float*)d_in[1];
  const float* prev0    = (const float*)d_in[2];
  const float* Wc1  = (const float*)d_in[3];
  const float* Wc2  = (const float*)d_in[4];
  const float* Wc3  = (const float*)d_in[5];
  const float* Wfw  = (const float*)d_in[6];
  const float* Wfwg = (const float*)d_in[7];
  const float* Wih1 = (const float*)d_in[8];
  const float* Whh1 = (const float*)d_in[9];
  const float* Wih2 = (const float*)d_in[10];
  const float* Whh2 = (const float*)d_in[11];
  const float* Wih3 = (const float*)d_in[12];
  const float* Whh3 = (const float*)d_in[13];
  const float* Wg1  = (const float*)d_in[14];
  const float* Wg2  = (const float*)d_in[15];
  const float* Wg3  = (const float*)d_in[16];
  const float* Wsg  = (const float*)d_in[17];
  const float* Wsd  = (const float*)d_in[18];
  const float* Wout = (const float*)d_in[19];
  float* outp = (float*)d_out;

  const size_t shmem = (size_t)SMEM_FLOATS * sizeof(float);
  (void)hipFuncSetAttribute((const void*)fargan_persistent,
                            hipFuncAttributeMaxDynamicSharedMemorySize, (int)shmem);
  fargan_persistent<<<dim3(4), dim3(NTHREADS), shmem, stream>>>(
      features, gfeat, prev0, Wc1, Wc2, Wc3, Wfw, Wfwg, Wih1, Whh1, Wih2, Whh2,
      Wih3, Whh3, Wg1, Wg2, Wg3, Wsg, Wsd, Wout, outp);
}